// IndomainOutdomainContrastiveLoss2_31224412242464
// MI455X (gfx1250) — compile-verified
//
#include <hip/hip_runtime.h>
#include <hip/hip_bf16.h>
#include <math.h>

// ---------------------------------------------------------------------------
// IndomainOutdomainContrastiveLoss on MI455X (gfx1250)
//
// loss = ( mean_i[lse_row_i(L)  - mrow_i /card_i]
//        + mean_j[lse_col_j(L)  - mcol_j /card_j]
//        + mean_i[lse_row_i(M)  - smrow_i/card2_i] ) / 3
// L = 100*img@txt^T (8192x8192), M = 100*img[:4096]@img[4096:]^T (4096x4096)
//
// Two-pass online softmax over recomputed f32 WMMA GEMM tiles (operands live
// in the 192MB L2, recompute is ~free; HBM sees ~67MB once = ~3us @23.3TB/s).
// Staging uses the CDNA5 async global->LDS engine (ASYNCcnt) with double
// buffering so v_wmma_f32_16x16x4_f32 overlaps the next K-panel's fill.
// LDS rows are padded to 36 dwords (144B) so fragment ds_loads are
// bank-conflict-free while keeping 16B alignment for async b128 writes.
// ---------------------------------------------------------------------------

typedef float v2f __attribute__((ext_vector_type(2)));
typedef float v8f __attribute__((ext_vector_type(8)));

#define N_FULL   8192
#define HALF_N   4096
#define DIMK     1024

#define BM       128   // block tile rows
#define BN       64    // block tile cols
#define KC       32    // K panel per stage
#define LDST     36    // padded LDS row stride (dwords): 144B, 16B-aligned,
                       // lane stride 36 dw -> 16 distinct bank groups
#define NSTAGE   (DIMK / KC)

// workspace layout (float units)
enum {
  OFF_ROW_MAX  = 0,
  OFF_COL_MAX  = OFF_ROW_MAX  + N_FULL,
  OFF_ROW_SUM  = OFF_COL_MAX  + N_FULL,
  OFF_COL_SUM  = OFF_ROW_SUM  + N_FULL,
  OFF_MROW     = OFF_COL_SUM  + N_FULL,
  OFF_MCOL     = OFF_MROW     + N_FULL,
  OFF_SROW_MAX = OFF_MCOL     + N_FULL,
  OFF_SCOL_MAX = OFF_SROW_MAX + HALF_N,
  OFF_SROW_SUM = OFF_SCOL_MAX + HALF_N,
  OFF_SCOL_SUM = OFF_SROW_SUM + HALF_N,
  OFF_SMROW    = OFF_SCOL_SUM + HALF_N,
  OFF_SMCOL    = OFF_SMROW    + HALF_N,
  OFF_COUNTS   = OFF_SMCOL    + HALF_N,   // int[1024]
  OFF_COUNTS_H = OFF_COUNTS   + 1024,     // int[1024]
  WS_TOTAL     = OFF_COUNTS_H + 1024
};

// low 32 bits of a generic pointer to __shared__ == LDS byte offset
__device__ __forceinline__ unsigned lds_lo(const void* p) {
  return (unsigned)(unsigned long long)p;
}

// async global->LDS 128-bit copy, GVS addressing (SGPR base + 32-bit VGPR off)
#define ASYNC_B128(ldsaddr, goff, base)                                   \
  asm volatile("global_load_async_to_lds_b128 %0, %1, %2 offset:0"        \
               :: "v"(ldsaddr), "v"(goff), "s"(base) : "memory")

// -------------------------------- init -------------------------------------
__global__ void init_ws_kernel(float* wsf) {
  int i = blockIdx.x * blockDim.x + threadIdx.x;
  if (i >= WS_TOTAL) return;
  float v = 0.0f;                                        // zero bits also zero the int counters
  if (i < OFF_ROW_SUM) v = -INFINITY;                    // row_max, col_max
  else if (i >= OFF_SROW_MAX && i < OFF_SROW_SUM) v = -INFINITY; // self maxes
  wsf[i] = v;
}

// ----------------------------- histogram -----------------------------------
__global__ void hist_kernel(const long long* __restrict__ tgt,
                            int* __restrict__ counts, int* __restrict__ counts_half) {
  int i = blockIdx.x * blockDim.x + threadIdx.x;
  if (i < N_FULL)  atomicAdd(&counts[(int)tgt[i]], 1);
  if (i < HALF_N)  atomicAdd(&counts_half[(int)tgt[i]], 1);
}

// --------------------- fused GEMM + contrastive epilogue -------------------
// C_tile = scale * A(M x K) @ B(N x K)^T   for one 128x64 tile per block.
// pass 1: atomic max_num row/col maxima + masked row/col sums
// pass 2: atomic add of exp(x - max) per row / per column
__global__ void __launch_bounds__(256)
gemm_scl_pass(const float* __restrict__ A, const float* __restrict__ B,
              const long long* __restrict__ tA, const long long* __restrict__ tB,
              const float* __restrict__ scale_ptr,
              float* __restrict__ row_max, float* __restrict__ col_max,
              float* __restrict__ row_sum, float* __restrict__ col_sum,
              float* __restrict__ mrow,    float* __restrict__ mcol,
              int pass)
{
  __shared__ __align__(16) float lA[2][BM * LDST];   // 2 x 18KB (padded)
  __shared__ __align__(16) float lB[2][BN * LDST];   // 2 x  9KB (padded)
  __shared__ int sTA[BM];
  __shared__ int sTB[BN];

  const int tid  = threadIdx.x;
  const int lane = tid & 31;
  const int w    = tid >> 5;              // 8 waves
  const int m0   = blockIdx.y * BM;
  const int n0   = blockIdx.x * BN;

  if (tid < BM)             sTA[tid]      = (int)tA[m0 + tid];
  else if (tid < BM + BN)   sTB[tid - BM] = (int)tB[n0 + tid - BM];

  const float scale = scale_ptr[0];
  const int wm   = w * 16;                // wave's 16-row strip, 4 N-subtiles
  const int half = lane >> 4;
  const int r16  = lane & 15;

  // per-thread async staging chunks: 4 float4 of A + 2 float4 of B per stage
  unsigned gA[4], dA[4], gB[2], dB[2];
#pragma unroll
  for (int q = 0; q < 4; ++q) {
    const int c = tid + 256 * q;                  // 1024 chunks of A panel
    const int m = c >> 3, k4 = (c & 7) << 2;      // 8 float4 per 32-wide row
    gA[q] = (unsigned)(((size_t)(m0 + m) * DIMK + k4) * 4);
    dA[q] = (unsigned)(m * LDST + k4) * 4;        // 16B-aligned (144*m + 16*j)
  }
#pragma unroll
  for (int q = 0; q < 2; ++q) {
    const int c = tid + 256 * q;                  // 512 chunks of B panel
    const int m = c >> 3, k4 = (c & 7) << 2;
    gB[q] = (unsigned)(((size_t)(n0 + m) * DIMK + k4) * 4);
    dB[q] = (unsigned)(m * LDST + k4) * 4;
  }
  const unsigned ldsA[2] = { lds_lo(&lA[0][0]), lds_lo(&lA[1][0]) };
  const unsigned ldsB[2] = { lds_lo(&lB[0][0]), lds_lo(&lB[1][0]) };

  v8f acc0 = {}, acc1 = {}, acc2 = {}, acc3 = {};

  // prologue: stage 0 into buffer 0
#pragma unroll
  for (int q = 0; q < 4; ++q) ASYNC_B128(ldsA[0] + dA[q], gA[q], A);
#pragma unroll
  for (int q = 0; q < 2; ++q) ASYNC_B128(ldsB[0] + dB[q], gB[q], B);

  for (int s = 0; s < NSTAGE; ++s) {
    if (s + 1 < NSTAGE) {
      // prefetch next K-panel into the other buffer, then wait for current 6
      // (async loads complete in order: cnt<=6 => the 6 of stage s are done)
      const unsigned koff = (unsigned)(s + 1) * (KC * 4);
      const int nb = (s + 1) & 1;
#pragma unroll
      for (int q = 0; q < 4; ++q) ASYNC_B128(ldsA[nb] + dA[q], gA[q] + koff, A);
#pragma unroll
      for (int q = 0; q < 2; ++q) ASYNC_B128(ldsB[nb] + dB[q], gB[q] + koff, B);
      asm volatile("s_wait_asynccnt 6" ::: "memory");
    } else {
      asm volatile("s_wait_asynccnt 0" ::: "memory");
    }
    __syncthreads();

    const float* pA = &lA[s & 1][0];
    const float* pB = &lB[s & 1][0];
#pragma unroll
    for (int k = 0; k < KC; k += 4) {
      // A frag 16x4: lanes 0-15 rows, VGPR0/1 = K{k,k+1}; upper half K{k+2,k+3}
      // B frag 4x16: lanes index N, VGPR0 = K{k | k+2}, VGPR1 = K{k+1 | k+3}
      const int ks = k + 2 * half;
      v2f a, b0, b1, b2, b3;
      a.x  = pA[(wm + r16) * LDST + ks];  a.y  = pA[(wm + r16) * LDST + ks + 1];
      b0.x = pB[(     r16) * LDST + ks];  b0.y = pB[(     r16) * LDST + ks + 1];
      b1.x = pB[(16 + r16) * LDST + ks];  b1.y = pB[(16 + r16) * LDST + ks + 1];
      b2.x = pB[(32 + r16) * LDST + ks];  b2.y = pB[(32 + r16) * LDST + ks + 1];
      b3.x = pB[(48 + r16) * LDST + ks];  b3.y = pB[(48 + r16) * LDST + ks + 1];
      acc0 = __builtin_amdgcn_wmma_f32_16x16x4_f32(false, a, false, b0, (short)0, acc0, false, false);
      acc1 = __builtin_amdgcn_wmma_f32_16x16x4_f32(false, a, false, b1, (short)0, acc1, false, false);
      acc2 = __builtin_amdgcn_wmma_f32_16x16x4_f32(false, a, false, b2, (short)0, acc2, false, false);
      acc3 = __builtin_amdgcn_wmma_f32_16x16x4_f32(false, a, false, b3, (short)0, acc3, false, false);
    }
    __syncthreads();   // all waves done reading this buffer before it refills
  }

  // ---------------- epilogue ----------------
  // C/D layout: VGPR r, lanes 0-15 -> row (wm+r), lanes 16-31 -> row (wm+r+8); N = lane&15
  const int nn = lane & 15;
  const v8f accs[4] = { acc0, acc1, acc2, acc3 };

#pragma unroll
  for (int t = 0; t < 4; ++t) {
    const v8f acc  = accs[t];
    const int colL = t * 16 + nn;
    const int col  = n0 + colL;
    const int tb   = sTB[colL];

    if (pass == 1) {
      float cmax = -INFINITY;
      float csum = 0.0f;
#pragma unroll
      for (int r = 0; r < 8; ++r) {
        const int rowL = wm + 8 * half + r;
        const int row  = m0 + rowL;
        const float v  = acc[r] * scale;
        float rm = v;
#pragma unroll
        for (int s = 1; s < 16; s <<= 1) rm = fmaxf(rm, __shfl_xor(rm, s, 32));
        const float mv = (sTA[rowL] == tb) ? v : 0.0f;
        float ms = mv;
#pragma unroll
        for (int s = 1; s < 16; s <<= 1) ms += __shfl_xor(ms, s, 32);
        if (nn == 0) {
          atomicMax(&row_max[row], rm);    // global_atomic_max_num_f32
          atomicAdd(&mrow[row], ms);
        }
        cmax = fmaxf(cmax, v);
        csum += mv;
      }
      cmax = fmaxf(cmax, __shfl_xor(cmax, 16, 32));
      csum += __shfl_xor(csum, 16, 32);
      if (half == 0) {
        atomicMax(&col_max[col], cmax);
        atomicAdd(&mcol[col], csum);
      }
    } else {
      const float cm = col_max[col];
      float csum = 0.0f;
#pragma unroll
      for (int r = 0; r < 8; ++r) {
        const int row = m0 + wm + 8 * half + r;
        const float v = acc[r] * scale;
        float e = __expf(v - row_max[row]);
        csum += __expf(v - cm);
#pragma unroll
        for (int s = 1; s < 16; s <<= 1) e += __shfl_xor(e, s, 32);
        if (nn == 0) atomicAdd(&row_sum[row], e);
      }
      csum += __shfl_xor(csum, 16, 32);
      if (half == 0) atomicAdd(&col_sum[col], csum);
    }
  }
}

// ------------------------------- finalize ----------------------------------
__global__ void __launch_bounds__(256)
finalize_kernel(const float* __restrict__ wsf,
                const long long* __restrict__ tgt,
                float* __restrict__ out)
{
  __shared__ float red[256];
  const float* row_max  = wsf + OFF_ROW_MAX;
  const float* col_max  = wsf + OFF_COL_MAX;
  const float* row_sum  = wsf + OFF_ROW_SUM;
  const float* col_sum  = wsf + OFF_COL_SUM;
  const float* mrow     = wsf + OFF_MROW;
  const float* mcol     = wsf + OFF_MCOL;
  const float* srow_max = wsf + OFF_SROW_MAX;
  const float* srow_sum = wsf + OFF_SROW_SUM;
  const float* smrow    = wsf + OFF_SMROW;
  const int* counts     = (const int*)(wsf + OFF_COUNTS);
  const int* counts_h   = (const int*)(wsf + OFF_COUNTS_H);

  float a1 = 0.0f, a2 = 0.0f, a3 = 0.0f;
  for (int i = threadIdx.x; i < N_FULL; i += 256) {
    const float c = (float)counts[(int)tgt[i]];
    a1 += row_max[i] + __logf(row_sum[i]) - mrow[i] / c;
    a2 += col_max[i] + __logf(col_sum[i]) - mcol[i] / c;
  }
  for (int i = threadIdx.x; i < HALF_N; i += 256) {
    const float c = (float)counts_h[(int)tgt[i]];
    a3 += srow_max[i] + __logf(srow_sum[i]) - smrow[i] / c;
  }
  float part = a1 * (1.0f / N_FULL) + a2 * (1.0f / N_FULL) + a3 * (1.0f / HALF_N);
  red[threadIdx.x] = part;
  __syncthreads();
  for (int s = 128; s > 0; s >>= 1) {
    if (threadIdx.x < s) red[threadIdx.x] += red[threadIdx.x + s];
    __syncthreads();
  }
  if (threadIdx.x == 0) out[0] = red[0] / 3.0f;
}

// ------------------------------- launcher ----------------------------------
extern "C" void kernel_launch(void* const* d_in, const int* in_sizes, int n_in,
                              void* d_out, int out_size, void* d_ws, size_t ws_size,
                              hipStream_t stream) {
  const float*     img   = (const float*)d_in[0];
  const float*     txt   = (const float*)d_in[1];
  const long long* tgt   = (const long long*)d_in[2];  // jnp.int64
  const float*     scale = (const float*)d_in[3];

  float* wsf       = (float*)d_ws;
  int*   counts    = (int*)(wsf + OFF_COUNTS);
  int*   counts_h  = (int*)(wsf + OFF_COUNTS_H);

  init_ws_kernel<<<(WS_TOTAL + 255) / 256, 256, 0, stream>>>(wsf);
  hist_kernel<<<(N_FULL + 255) / 256, 256, 0, stream>>>(tgt, counts, counts_h);

  const dim3 blk(256);
  const dim3 gL(N_FULL / BN, N_FULL / BM);
  const dim3 gS(HALF_N / BN, HALF_N / BM);
  const float* imgB = img + (size_t)HALF_N * DIMK;

  // pass 1: maxima + masked sums
  gemm_scl_pass<<<gL, blk, 0, stream>>>(img, txt, tgt, tgt, scale,
      wsf + OFF_ROW_MAX,  wsf + OFF_COL_MAX,  wsf + OFF_ROW_SUM,  wsf + OFF_COL_SUM,
      wsf + OFF_MROW,     wsf + OFF_MCOL, 1);
  gemm_scl_pass<<<gS, blk, 0, stream>>>(img, imgB, tgt, tgt, scale,
      wsf + OFF_SROW_MAX, wsf + OFF_SCOL_MAX, wsf + OFF_SROW_SUM, wsf + OFF_SCOL_SUM,
      wsf + OFF_SMROW,    wsf + OFF_SMCOL, 1);

  // pass 2: sum of exp(x - max)
  gemm_scl_pass<<<gL, blk, 0, stream>>>(img, txt, tgt, tgt, scale,
      wsf + OFF_ROW_MAX,  wsf + OFF_COL_MAX,  wsf + OFF_ROW_SUM,  wsf + OFF_COL_SUM,
      wsf + OFF_MROW,     wsf + OFF_MCOL, 2);
  gemm_scl_pass<<<gS, blk, 0, stream>>>(img, imgB, tgt, tgt, scale,
      wsf + OFF_SROW_MAX, wsf + OFF_SCOL_MAX, wsf + OFF_SROW_SUM, wsf + OFF_SCOL_SUM,
      wsf + OFF_SMROW,    wsf + OFF_SMCOL, 2);

  finalize_kernel<<<1, 256, 0, stream>>>(wsf, tgt, (float*)d_out);
}